// GCN_76501957476420
// MI455X (gfx1250) — compile-verified
//
#include <hip/hip_runtime.h>
#include <hip/hip_bf16.h>

typedef __attribute__((ext_vector_type(2))) float v2f;
typedef __attribute__((ext_vector_type(8))) float v8f;

#define HDIM 128
#define FIN  11

// ---------------- small utility kernels ----------------

__global__ void fill_f32(float* p, float v, int n) {
    int i = blockIdx.x * blockDim.x + threadIdx.x;
    if (i < n) p[i] = v;
}

__global__ void deg_acc(const int* __restrict__ dst, float* __restrict__ deg, int E) {
    int e = blockIdx.x * blockDim.x + threadIdx.x;
    if (e < E) unsafeAtomicAdd(&deg[dst[e]], 1.0f);
}

__global__ void deg_to_dinv(float* deg, int n) {
    int i = blockIdx.x * blockDim.x + threadIdx.x;
    if (i < n) deg[i] = __frsqrt_rn(deg[i]);   // deg >= 1 always (self-loops)
}

__global__ void norm_k(const int* __restrict__ src, const int* __restrict__ dst,
                       const float* __restrict__ dinv, float* __restrict__ norm,
                       int E, int Etot) {
    int e = blockIdx.x * blockDim.x + threadIdx.x;
    if (e >= Etot) return;
    if (e < E) norm[e] = dinv[src[e]] * dinv[dst[e]];
    else {
        float d = dinv[e - E];
        norm[e] = d * d;
    }
}

// out[n,c] = b[c]  (bias pre-seeded into scatter accumulator)
__global__ void init_bias(float* __restrict__ out, const float* __restrict__ b, int total) {
    int i = blockIdx.x * blockDim.x + threadIdx.x;
    if (i < total) out[i] = b[i & (HDIM - 1)];
}

__global__ void relu_k(float* h, int total) {
    int i = blockIdx.x * blockDim.x + threadIdx.x;
    if (i < total) h[i] = fmaxf(h[i], 0.0f);
}

// ---------------- skinny input GEMM: [N,11] @ [11,128] ----------------

__global__ void gemm_in(const float* __restrict__ x, const float* __restrict__ W1,
                        float* __restrict__ out, int N) {
    int i = blockIdx.x * blockDim.x + threadIdx.x;
    if (i >= N * HDIM) return;
    int n = i >> 7, c = i & (HDIM - 1);
    float acc = 0.0f;
#pragma unroll
    for (int k = 0; k < FIN; ++k)
        acc = fmaf(x[n * FIN + k], W1[k * HDIM + c], acc);
    out[i] = acc;
}

// ---------------- WMMA FP32 GEMM: [N,128] @ [128,128] ----------------
// Block = 256 threads (8 waves). Block handles a 16-row strip of A (staged in
// LDS); wave w computes the 16x16 output tile at columns [16w, 16w+16).
// K-loop: 32 x V_WMMA_F32_16X16X4_F32.
// Fragment layouts per CDNA5 ISA 7.12.2 (wave32):
//   A 16x4:  lanes 0-15 M=lane {K=k,k+1}; lanes 16-31 M=lane-16 {K=k+2,k+3}
//   B 4x16:  lanes 0-15 N=lane {K=k,k+1}; lanes 16-31 N=lane-16 {K=k+2,k+3}
//   C 16x16: VGPR j -> (M = j + 8*(lane>=16), N = lane%16)
__global__ void gemm128_wmma(const float* __restrict__ A, const float* __restrict__ W,
                             float* __restrict__ Out, int N) {
    __shared__ float As[16 * HDIM];
    const int tid  = threadIdx.x;
    const int row0 = blockIdx.x * 16;

    for (int i = tid; i < 16 * HDIM; i += 256)
        As[i] = A[(size_t)row0 * HDIM + i];
    __syncthreads();

    const int wave = tid >> 5;
    const int lane = tid & 31;
    const int half = lane >> 4;     // 0: K pair (k,k+1)   1: K pair (k+2,k+3)
    const int l    = lane & 15;
    const int col0 = wave * 16;

    v8f c = {};
#pragma unroll 8
    for (int k = 0; k < HDIM; k += 4) {
        v2f a, b;
        a.x = As[l * HDIM + k + half * 2];
        a.y = As[l * HDIM + k + half * 2 + 1];
        const float* bp = W + (size_t)(k + half * 2) * HDIM + col0 + l;
        b.x = bp[0];
        b.y = bp[HDIM];
        c = __builtin_amdgcn_wmma_f32_16x16x4_f32(false, a, false, b,
                                                  (short)0, c, false, false);
    }

    float* out = Out + (size_t)row0 * HDIM + col0;
#pragma unroll
    for (int j = 0; j < 8; ++j)
        out[(size_t)(j + 8 * half) * HDIM + l] = c[j];
}

// ---------------- edge aggregation (normalized scatter-add) ----------------
// One wave per edge; each lane owns 4 channels (32*4 = 128). float4 gather
// from source row (L2-resident), 4x global_atomic_add_f32 to dest row.

__global__ void agg_edges(const float* __restrict__ hin, float* __restrict__ hout,
                          const int* __restrict__ src, const int* __restrict__ dst,
                          const float* __restrict__ norm, int E, int Etot) {
    long long gid = (long long)blockIdx.x * blockDim.x + threadIdx.x;
    int e = (int)(gid >> 5);
    int lane = (int)(gid & 31);
    if (e >= Etot) return;
    int s, d;
    if (e < E) { s = src[e]; d = dst[e]; }
    else       { s = d = e - E; }                  // self-loop
    float w = norm[e];
    const float4 v = *(const float4*)(hin + (size_t)s * HDIM + lane * 4);
    float* o = hout + (size_t)d * HDIM + lane * 4;
    unsafeAtomicAdd(o + 0, v.x * w);
    unsafeAtomicAdd(o + 1, v.y * w);
    unsafeAtomicAdd(o + 2, v.z * w);
    unsafeAtomicAdd(o + 3, v.w * w);
}

// ---------------- segment-mean pooling + final linear ----------------

__global__ void pool_acc(const float* __restrict__ h, const int* __restrict__ batch,
                         float* __restrict__ pooled, float* __restrict__ cnt, int N) {
    long long gid = (long long)blockIdx.x * blockDim.x + threadIdx.x;
    int n = (int)(gid >> 5);
    int lane = (int)(gid & 31);
    if (n >= N) return;
    int g = batch[n];
    const float4 v = *(const float4*)(h + (size_t)n * HDIM + lane * 4);
    float* o = pooled + (size_t)g * HDIM + lane * 4;
    unsafeAtomicAdd(o + 0, v.x);
    unsafeAtomicAdd(o + 1, v.y);
    unsafeAtomicAdd(o + 2, v.z);
    unsafeAtomicAdd(o + 3, v.w);
    if (lane == 0) unsafeAtomicAdd(&cnt[g], 1.0f);
}

__global__ void final_k(const float* __restrict__ pooled, const float* __restrict__ cnt,
                        const float* __restrict__ Wl, const float* __restrict__ bl,
                        float* __restrict__ out, int G) {
    int g = blockIdx.x * blockDim.x + threadIdx.x;
    if (g >= G) return;
    float c = fmaxf(cnt[g], 1.0f);
    float acc = 0.0f;
#pragma unroll 8
    for (int k = 0; k < HDIM; ++k)
        acc = fmaf(pooled[(size_t)g * HDIM + k], Wl[k], acc);
    out[g] = acc / c + bl[0];
}

// ---------------- host orchestration ----------------

extern "C" void kernel_launch(void* const* d_in, const int* in_sizes, int n_in,
                              void* d_out, int out_size, void* d_ws, size_t ws_size,
                              hipStream_t stream) {
    const float* x    = (const float*)d_in[0];
    const int*   ei   = (const int*)d_in[1];     // [2,E] flat: src then dst
    const int*   batch= (const int*)d_in[2];
    const float* W1   = (const float*)d_in[3];
    const float* b1   = (const float*)d_in[4];
    const float* W2   = (const float*)d_in[5];
    const float* b2   = (const float*)d_in[6];
    const float* W3   = (const float*)d_in[7];
    const float* b3   = (const float*)d_in[8];
    const float* Wl   = (const float*)d_in[9];
    const float* bl   = (const float*)d_in[10];
    float* out = (float*)d_out;

    const int N = in_sizes[0] / FIN;
    const int E = in_sizes[1] / 2;
    const int T = in_sizes[10];
    const int G = out_size / T;
    const int Etot = E + N;
    const int NH = N * HDIM;

    const int* src = ei;
    const int* dst = ei + E;

    // workspace carve-up (256B aligned)
    char* ws = (char*)d_ws;
    size_t off = 0;
    auto carve = [&](size_t bytes) -> char* {
        char* p = ws + off;
        off = (off + bytes + 255) & ~(size_t)255;
        return p;
    };
    float* hA     = (float*)carve((size_t)NH * 4);
    float* hB     = (float*)carve((size_t)NH * 4);
    float* dinv   = (float*)carve((size_t)N * 4);
    float* norm   = (float*)carve((size_t)Etot * 4);
    float* pooled = (float*)carve((size_t)G * HDIM * 4);
    float* cnt    = (float*)carve((size_t)G * 4);
    (void)ws_size;

    const int B = 256;
    auto blk = [&](long long n) { return (unsigned)((n + B - 1) / B); };

    // degree / normalization
    fill_f32<<<blk(N), B, 0, stream>>>(dinv, 1.0f, N);           // self-loop count
    deg_acc<<<blk(E), B, 0, stream>>>(dst, dinv, E);
    deg_to_dinv<<<blk(N), B, 0, stream>>>(dinv, N);
    norm_k<<<blk(Etot), B, 0, stream>>>(src, dst, dinv, norm, E, Etot);

    const long long edgeThreads = (long long)Etot * 32;
    const long long poolThreads = (long long)N * 32;

    // layer 1: x@W1 -> hB ; aggregate -> hA (+b1) ; relu
    gemm_in<<<blk(NH), B, 0, stream>>>(x, W1, hB, N);
    init_bias<<<blk(NH), B, 0, stream>>>(hA, b1, NH);
    agg_edges<<<blk(edgeThreads), B, 0, stream>>>(hB, hA, src, dst, norm, E, Etot);
    relu_k<<<blk(NH), B, 0, stream>>>(hA, NH);

    // layer 2: hA@W2 -> hB (WMMA) ; aggregate -> hA (+b2) ; relu
    gemm128_wmma<<<N / 16, B, 0, stream>>>(hA, W2, hB, N);
    init_bias<<<blk(NH), B, 0, stream>>>(hA, b2, NH);
    agg_edges<<<blk(edgeThreads), B, 0, stream>>>(hB, hA, src, dst, norm, E, Etot);
    relu_k<<<blk(NH), B, 0, stream>>>(hA, NH);

    // layer 3: hA@W3 -> hB (WMMA) ; aggregate -> hA (+b3), no relu
    gemm128_wmma<<<N / 16, B, 0, stream>>>(hA, W3, hB, N);
    init_bias<<<blk(NH), B, 0, stream>>>(hA, b3, NH);
    agg_edges<<<blk(edgeThreads), B, 0, stream>>>(hB, hA, src, dst, norm, E, Etot);

    // pooling + final linear
    fill_f32<<<blk(G * HDIM), B, 0, stream>>>(pooled, 0.0f, G * HDIM);
    fill_f32<<<blk(G), B, 0, stream>>>(cnt, 0.0f, G);
    pool_acc<<<blk(poolThreads), B, 0, stream>>>(hA, batch, pooled, cnt, N);
    final_k<<<blk(G), B, 0, stream>>>(pooled, cnt, Wl, bl, out, G);
}